// FastElasticInteractionEnergyLoss_51204600103263
// MI455X (gfx1250) — compile-verified
//
#include <hip/hip_runtime.h>
#include <math.h>

typedef __attribute__((ext_vector_type(2))) float v2f;
typedef __attribute__((ext_vector_type(8))) float v8f;

namespace {
constexpr int   B_    = 32;
constexpr int   L_    = 8;
constexpr int   P_    = 1024;
constexpr int   S_    = P_ - 1;              // 1023 segments
constexpr int   BL_   = B_ * L_;             // 256 (b,l) lanes
constexpr int   NT_   = BL_ * S_;            // total segments (line kernel)
constexpr int   TILES_ = (S_ + 15) / 16;     // 64 tiles per dim
constexpr int   LPT_   = TILES_ * 32;        // v2f elements per (b,l) per array (2048)
constexpr int   NFMT_  = BL_ * LPT_;         // v2f elements per array (524288)
constexpr float EPSF   = 1e-8f;
constexpr float PIF    = 3.14159265358979323846f;
constexpr float FLTMAX = 3.402823466e38f;
// scales (1/B folded in)
constexpr float LINE_SCALE = 1.0f / (8.0f * PIF * (float)B_);
constexpr float SELF_SCALE = 1.0f / (8.0f * PIF * (float)B_);
// lane_loss has  -2*ALPHA*e_int  with  e_int = sum(-dots/(4*pi*r))
//   => contribution = +dots / (2*pi*r)    (ALPHA = 1)
constexpr float INT_SCALE  = 1.0f / (2.0f * PIF * (float)B_);
} // namespace

// ---------------------------------------------------------------------------
// Line-length term: sum |dl| over all segments of pred+gt, scaled by 1/(8*pi*B).
// ---------------------------------------------------------------------------
__global__ __launch_bounds__(256)
void feiel_line_kernel(const float* __restrict__ pred,
                       const float* __restrict__ gt,
                       float* __restrict__ out)
{
    const int idx = blockIdx.x * 256 + threadIdx.x;
    float lineSum = 0.0f;
    if (idx < NT_) {
        const int s  = idx % S_;
        const int bl = idx / S_;
        const size_t base = ((size_t)bl * P_ + s) * 2;
        {
            const float dx = pred[base + 2] - pred[base + 0];
            const float dy = pred[base + 3] - pred[base + 1];
            const float ss = dx * dx + dy * dy;
            const float sc = __builtin_amdgcn_sqrtf(ss) *
                             __builtin_amdgcn_rsqf(ss + EPSF);
            const float dlx = dx * sc, dly = dy * sc;
            lineSum += __builtin_amdgcn_sqrtf(dlx * dlx + dly * dly);
        }
        {
            const float dx = gt[base + 2] - gt[base + 0];
            const float dy = gt[base + 3] - gt[base + 1];
            const float ss = dx * dx + dy * dy;
            const float sc = __builtin_amdgcn_sqrtf(ss) *
                             __builtin_amdgcn_rsqf(ss + EPSF);
            const float dlx = dx * sc, dly = dy * sc;
            lineSum += __builtin_amdgcn_sqrtf(dlx * dlx + dly * dly);
        }
    }
    __shared__ float red[256];
    red[threadIdx.x] = lineSum;
    __syncthreads();
    for (int s = 128; s > 0; s >>= 1) {
        if ((int)threadIdx.x < s) red[threadIdx.x] += red[threadIdx.x + s];
        __syncthreads();
    }
    if (threadIdx.x == 0) atomicAdd(out, red[0] * LINE_SCALE);
}

// ---------------------------------------------------------------------------
// Format: build WMMA-ready per-lane operand vectors, one thread per
// (bl, tile, lane), for both pred and gt.
// ws layout, 6 arrays of NFMT_ v2f (4 MB each, ~24 MB total, L2-resident):
//   [0]=dl_p  [1]=aR_p  [2]=bR_p  [3]=dl_g  [4]=aR_g  [5]=bR_g
// dl  : lanes 0-15 = (dx,dy) (zeroed if row OOB), lanes 16-31 = 0 (K2/K3 pad)
//       (same layout serves both the A operand and the B operand)
// aR  : row i = (mx,my | |m|^2,1)   -> r^2 GEMM A side
// bR  : col j = (-2mx,-2my | 1,|m|^2) -> r^2 GEMM B side
// ---------------------------------------------------------------------------
__global__ __launch_bounds__(256)
void feiel_format_kernel(const float* __restrict__ pred,
                         const float* __restrict__ gt,
                         v2f* __restrict__ ws)
{
    const int gid = blockIdx.x * 256 + threadIdx.x;   // 0 .. NFMT_-1 (exact)
    const int lane = gid & 31;
    const int tile = (gid >> 5) & (TILES_ - 1);
    const int bl   = gid >> 11;                       // /2048
    const int half = lane >> 4;
    const int l16  = lane & 15;
    const int iv   = tile * 16 + l16;
    const int inR  = (iv < S_) ? 1 : 0;
    const int si   = inR ? iv : (S_ - 1);
    const size_t base = ((size_t)bl * P_ + si) * 2;
    const float dlMask = (half == 0 && inR) ? 1.0f : 0.0f;
    const int   oidx = gid;   // bl*2048 + tile*32 + lane

#pragma unroll
    for (int side = 0; side < 2; ++side) {
        const float* __restrict__ src = side ? gt : pred;
        const float x0 = src[base + 0], y0 = src[base + 1];
        const float x1 = src[base + 2], y1 = src[base + 3];
        const float dx = x1 - x0, dy = y1 - y0;
        const float ss = dx * dx + dy * dy;
        const float sc = __builtin_amdgcn_sqrtf(ss) *
                         __builtin_amdgcn_rsqf(ss + EPSF);
        const float dlx = dx * sc, dly = dy * sc;
        const float mx = 0.5f * (x0 + x1), my = 0.5f * (y0 + y1);
        const float sm = mx * mx + my * my;

        v2f dl; dl.x = dlx * dlMask; dl.y = dly * dlMask;
        v2f aR; aR.x = half ? sm   : mx;            aR.y = half ? 1.0f : my;
        v2f bR; bR.x = half ? 1.0f : (-2.0f * mx);  bR.y = half ? sm   : (-2.0f * my);

        v2f* __restrict__ dst = ws + (size_t)(side * 3) * NFMT_;
        dst[0 * (size_t)NFMT_ + oidx] = dl;
        dst[1 * (size_t)NFMT_ + oidx] = aR;
        dst[2 * (size_t)NFMT_ + oidx] = bR;
    }
}

// ---------------------------------------------------------------------------
// Pairwise energies via WMMA 16x16x4 f32 on preformatted operands.
// blockIdx.x = (b,l); blockIdx.y = mode (0=self_p, 1=self_g, 2=interaction)
// Each wave owns output columns tj (B operands loaded once per column);
// inner ti loop: 2 coalesced global_load_b64 + 2 WMMA + branchless epilogue.
// Self modes pair columns (c, 63-c) => perfectly balanced.
// OOB pairs contribute exactly 0 because their dl operands were zeroed.
// v8f accumulator => 8 independent dependency chains.
// ---------------------------------------------------------------------------
__global__ __launch_bounds__(256)
void feiel_pair_kernel(const v2f* __restrict__ ws,
                       float* __restrict__ out)
{
    const int bl   = blockIdx.x;
    const int mode = blockIdx.y;            // 0,1,2
    const int wave = threadIdx.x >> 5;      // 8 waves
    const int lane = threadIdx.x & 31;
    const int half = lane >> 4;
    const int l16  = lane & 15;
    const int n    = l16;                   // C/D column for this lane
    const int mb   = half * 8;              // C/D row base for this lane

    // array bases (v2f elements)
    const v2f* dl_p = ws + 0 * (size_t)NFMT_;
    const v2f* aR_p = ws + 1 * (size_t)NFMT_;
    const v2f* bR_p = ws + 2 * (size_t)NFMT_;
    const v2f* dl_g = ws + 3 * (size_t)NFMT_;
    const v2f* aR_g = ws + 4 * (size_t)NFMT_;
    const v2f* bR_g = ws + 5 * (size_t)NFMT_;

    // A side: pred for modes 0,2; gt for mode 1.  B side: pred mode 0; gt modes 1,2.
    const size_t blLane = (size_t)bl * LPT_ + lane;
    const v2f* __restrict__ Adl = ((mode == 1) ? dl_g : dl_p) + blLane;
    const v2f* __restrict__ AaR = ((mode == 1) ? aR_g : aR_p) + blLane;
    const v2f* __restrict__ Bdl = ((mode == 0) ? dl_p : dl_g) + blLane;
    const v2f* __restrict__ BbR = ((mode == 0) ? bR_p : bR_g) + blLane;

    const int   fullTile = (mode == 2) ? 1 : 0;
    const float scale    = fullTile ? INT_SCALE : SELF_SCALE;

    v8f accv = {};

    // Process one column of 16x16 tiles (fixed tj, ti = 0..tiEnd-1).
    auto doColumn = [&](int tj) {
        const v2f b_dl = Bdl[(size_t)tj * 32];
        const v2f b_r  = BbR[(size_t)tj * 32];
        const int tiEnd = fullTile ? TILES_ : (tj + 1);
        for (int ti = 0; ti < tiEnd; ++ti) {
            const v2f a_dl = Adl[(size_t)ti * 32];
            const v2f a_r  = AaR[(size_t)ti * 32];

            v8f zero = {};
            v8f dots = __builtin_amdgcn_wmma_f32_16x16x4_f32(
                false, a_dl, false, b_dl, (short)0, zero, false, false);
            v8f r2 = __builtin_amdgcn_wmma_f32_16x16x4_f32(
                false, a_r, false, b_r, (short)0, zero, false, false);

            if (!fullTile && ti == tj) {
                // diagonal tile: keep only i < j  <=>  (mb+v) < n
#pragma unroll
                for (int v = 0; v < 8; ++v) {
                    const float d2 = __builtin_amdgcn_fmed3f(r2[v], 0.0f, FLTMAX);
                    const float r  = __builtin_amdgcn_sqrtf(d2) + EPSF;
                    const float c  = dots[v] * __builtin_amdgcn_rcpf(r);
                    accv[v] += ((mb + v) < n) ? c : 0.0f;
                }
            } else {
#pragma unroll
                for (int v = 0; v < 8; ++v) {
                    const float d2 = __builtin_amdgcn_fmed3f(r2[v], 0.0f, FLTMAX);
                    const float r  = __builtin_amdgcn_sqrtf(d2) + EPSF;
                    accv[v] += dots[v] * __builtin_amdgcn_rcpf(r);
                }
            }
        }
    };

    if (fullTile) {
        for (int c = wave; c < TILES_; c += 8) doColumn(c);
    } else {
        // pair columns (c, 63-c): each pair = 65 tiles => perfect balance
        for (int c = wave; c < TILES_ / 2; c += 8) {
            doColumn(c);
            doColumn(TILES_ - 1 - c);
        }
    }

    float acc = ((accv[0] + accv[1]) + (accv[2] + accv[3])) +
                ((accv[4] + accv[5]) + (accv[6] + accv[7]));

    // wave32 reduction, then one atomic per wave
    for (int off = 16; off > 0; off >>= 1)
        acc += __shfl_down(acc, off, 32);
    if (lane == 0) atomicAdd(out, acc * scale);
}

extern "C" void kernel_launch(void* const* d_in, const int* in_sizes, int n_in,
                              void* d_out, int out_size, void* d_ws, size_t ws_size,
                              hipStream_t stream) {
    (void)in_sizes; (void)n_in; (void)ws_size;
    const float* pred = (const float*)d_in[0];
    const float* gt   = (const float*)d_in[1];
    float* out = (float*)d_out;
    v2f*   ws  = (v2f*)d_ws;

    hipMemsetAsync(out, 0, (size_t)out_size * sizeof(float), stream);

    feiel_line_kernel<<<(NT_ + 255) / 256, 256, 0, stream>>>(pred, gt, out);
    feiel_format_kernel<<<NFMT_ / 256, 256, 0, stream>>>(pred, gt, ws);

    dim3 grid(BL_, 3);
    feiel_pair_kernel<<<grid, 256, 0, stream>>>(ws, out);
}